// EmbeddingLayer_69097433858479
// MI455X (gfx1250) — compile-verified
//
#include <hip/hip_runtime.h>
#include <stdint.h>

// Problem constants (from reference)
#define B_TOTAL   16384
#define N_SPARSE  8
#define N_SEQ     2
#define N_DENSE   4
#define VOCAB     100000
#define D         64
#define L         50
#define OUT_COLS  ((N_SPARSE + N_SEQ) * D + N_DENSE)   // 644

#define ROWS_PER_BLOCK 16
#define THREADS        256          // 16 lanes per row, each lane owns 4 columns (float4)
#define IDS_PER_BLOCK  (ROWS_PER_BLOCK * N_SEQ * L)    // 1600 ints = 6400 B LDS

// ---- CDNA5 async global->LDS path (guarded; falls back to plain loads) ----
#if defined(__gfx1250__) && __has_builtin(__builtin_amdgcn_global_load_async_to_lds_b128)
#define USE_ASYNC 1
typedef int v4i __attribute__((vector_size(16)));            // matches builtin's vector param
typedef __attribute__((address_space(1))) v4i g_v4i;         // global (prints as __device__)
typedef __attribute__((address_space(3))) v4i s_v4i;         // LDS
#else
#define USE_ASYNC 0
#endif

__global__ __launch_bounds__(THREADS) void
embedding_gather_pool_kernel(const int*   __restrict__ sparse_ids,   // [B, 8]
                             const int*   __restrict__ seq_ids,      // [B, 2, 50]
                             const float* __restrict__ dense,        // [B, 4]
                             const float* __restrict__ tsp,          // [8, VOCAB, 64]
                             const float* __restrict__ tsq,          // [2, VOCAB, 64]
                             float*       __restrict__ out)          // [B, 644]
{
    __shared__ __align__(16) int ids[IDS_PER_BLOCK];

    const int tid = threadIdx.x;
    const int b0  = blockIdx.x * ROWS_PER_BLOCK;

    // ---- Stage this block's seq ids (contiguous 1600 ints) into LDS ----
    const int* gids = seq_ids + (size_t)b0 * (N_SEQ * L);
#if USE_ASYNC
    for (int i = tid; i < IDS_PER_BLOCK / 4; i += THREADS) {
        __builtin_amdgcn_global_load_async_to_lds_b128(
            (g_v4i*)(uintptr_t)(gids + 4 * i),
            (s_v4i*)(uint32_t)(uintptr_t)(&ids[4 * i]),
            /*imm offset*/0, /*cpol*/0);
    }
#if __has_builtin(__builtin_amdgcn_s_wait_asynccnt)
    __builtin_amdgcn_s_wait_asynccnt(0);
#else
    asm volatile("s_wait_asynccnt 0" ::: "memory");
#endif
#else
    for (int i = tid; i < IDS_PER_BLOCK; i += THREADS) ids[i] = gids[i];
#endif
    __syncthreads();

    const int r   = tid >> 4;            // row within block: 0..15
    const int c   = (tid & 15) << 2;     // column offset in floats: 0,4,...,60
    const int row = b0 + r;
    float* __restrict__ orow = out + (size_t)row * OUT_COLS;

    // ---- Prefetch the 100 seq-embedding rows for this batch row (one lane) ----
#if defined(__gfx1250__)
    if ((tid & 15) == 0) {
        #pragma unroll
        for (int f = 0; f < N_SEQ; ++f) {
            #pragma unroll 10
            for (int l = 0; l < L; ++l) {
                const int id = ids[(r * N_SEQ + f) * L + l];
                const int idc = (id == -1) ? 0 : id;
                __builtin_prefetch(tsq + ((size_t)f * VOCAB + idc) * D, 0, 3);
            }
        }
    }
#endif

    // ---- Sparse gathers: 8 fully coalesced b128 loads, written straight out ----
    int sid[N_SPARSE];
    #pragma unroll
    for (int t = 0; t < N_SPARSE; ++t)
        sid[t] = sparse_ids[(size_t)row * N_SPARSE + t];
    #pragma unroll
    for (int t = 0; t < N_SPARSE; ++t) {
        const float4 v = *(const float4*)(tsp + ((size_t)t * VOCAB + sid[t]) * D + c);
        *(float4*)(orow + t * D + c) = v;
    }

    // ---- Sequence pooling: masked sum (f=0), masked mean (f=1) ----
    #pragma unroll
    for (int f = 0; f < N_SEQ; ++f) {
        const int* lid = &ids[(r * N_SEQ + f) * L];
        float ax = 0.f, ay = 0.f, az = 0.f, aw = 0.f, cnt = 0.f;
        #pragma unroll 10
        for (int l = 0; l < L; ++l) {
            const int id  = lid[l];
            const float m = (id != -1) ? 1.0f : 0.0f;   // reference mask (ids != -1)
            const int idc = (id == -1) ? 0 : id;        // keep load in-bounds
            const float4 v = *(const float4*)(tsq + ((size_t)f * VOCAB + idc) * D + c);
            ax += m * v.x; ay += m * v.y; az += m * v.z; aw += m * v.w;
            cnt += m;
        }
        float4 o;
        if (f == 0) {
            o.x = ax; o.y = ay; o.z = az; o.w = aw;
            *(float4*)(orow + N_SPARSE * D + c) = o;
        } else {
            const float den = cnt + 1e-16f;
            o.x = ax / den; o.y = ay / den; o.z = az / den; o.w = aw / den;
            *(float4*)(orow + (N_SPARSE + 1) * D + c) = o;
        }
    }

    // ---- Dense passthrough: one lane per row copies 4 floats ----
    if ((tid & 15) == 0) {
        const float4 dv = *(const float4*)(dense + (size_t)row * N_DENSE);
        *(float4*)(orow + (N_SPARSE + N_SEQ) * D) = dv;
    }
}

extern "C" void kernel_launch(void* const* d_in, const int* in_sizes, int n_in,
                              void* d_out, int out_size, void* d_ws, size_t ws_size,
                              hipStream_t stream) {
    (void)in_sizes; (void)n_in; (void)d_ws; (void)ws_size; (void)out_size;
    const int*   sparse_ids = (const int*)  d_in[0];   // [B, 8]       int32
    const int*   seq_ids    = (const int*)  d_in[1];   // [B, 2, 50]   int32
    const float* dense      = (const float*)d_in[2];   // [B, 4]       fp32
    const float* tsp        = (const float*)d_in[3];   // [8, VOCAB, 64] fp32
    const float* tsq        = (const float*)d_in[4];   // [2, VOCAB, 64] fp32
    float*       out        = (float*)      d_out;     // [B, 644]     fp32

    const int blocks = B_TOTAL / ROWS_PER_BLOCK;       // 1024
    embedding_gather_pool_kernel<<<blocks, THREADS, 0, stream>>>(
        sparse_ids, seq_ids, dense, tsp, tsq, out);
}